// Fbank_66039417143329
// MI455X (gfx1250) — compile-verified
//
#include <hip/hip_runtime.h>
#include <math.h>

// ---------------------------------------------------------------------------
// Fbank on gfx1250: rFFT power spectrum expressed as a dense real-DFT GEMM
// executed on the fp32 WMMA pipe (v_wmma_f32_16x16x4_f32), mel projection on
// VALU with analytic band limits. One workgroup (256 thr = 8 wave32) handles
// 16 frames of one batch row.
// ---------------------------------------------------------------------------

typedef float v2f __attribute__((ext_vector_type(2)));
typedef float v8f __attribute__((ext_vector_type(8)));

#define BATCH       32
#define TSAMP       160000
#define FRAME_LEN   400
#define FRAME_SHIFT 160
#define NFFT        512
#define NBINS       256          // bin 256 hits a zero mel column -> skip it
#define KMAX        400          // samples >= 400 are zero -> skip in GEMM K
#define NMEL        80
#define NFRAMES     998          // 1 + (160000-400)/160
#define FR_TILE     16
#define NTILES      63           // ceil(998/16)
#define ROWPAD      516          // frames row stride (floats): breaks bank conflicts
#define SPECPAD     260
#define BINW        31.25f       // SR / NFFT

__global__ __launch_bounds__(256) void fbank_kernel(const float* __restrict__ x,
                                                    float* __restrict__ out)
{
    __shared__ float2 trig[NFFT];                  // (cos, sin)(2*pi*m/512)
    __shared__ float  melbin[NBINS];               // mel(31.25 * k)
    __shared__ float  wnd[FRAME_LEN];              // Hamming window
    __shared__ float  frames[FR_TILE * ROWPAD];    // windowed frame rows
    __shared__ float  spec[FR_TILE * SPECPAD];     // power spectrum

    const int tid  = threadIdx.x;
    const int lane = tid & 31;
    const int wave = tid >> 5;

    const int batch = blockIdx.x / NTILES;
    const int tile  = blockIdx.x % NTILES;
    const int fr0   = tile * FR_TILE;

    // ---- phase 0: LDS tables (trig is exact per element: idx = (bin*n)&511)
    {
        const float twopi = 6.28318530717958647692f;
        #pragma unroll
        for (int e = 0; e < 2; ++e) {
            int m = tid + 256 * e;
            float a = twopi * (float)m * (1.0f / 512.0f);
            trig[m] = make_float2(cosf(a), sinf(a));
        }
        // melbin[k] = 1127*log1p(fft_bin_width*k/700)
        melbin[tid & 255] = 1127.0f * log1pf(BINW * (float)(tid & 255) * (1.0f / 700.0f));
        for (int m = tid; m < FRAME_LEN; m += 256)
            wnd[m] = 0.54f - 0.46f * cosf(6.28318530717958647692f *
                                          (float)m * (1.0f / 399.0f));
    }
    __syncthreads();

    // ---- phase 1: frame extraction / mean removal / pre-emphasis / window
    {
        const int frl = 2 * wave + (lane >> 4);    // local frame 0..15
        const int j   = lane & 15;                 // 16 lanes per frame
        const int frg = fr0 + frl;
        const bool valid = (frg < NFRAMES);
        const float* xf = x + (size_t)batch * TSAMP + (size_t)frg * FRAME_SHIFT;

        float vals[25];
        float s = 0.0f;
        if (valid) {
            #pragma unroll
            for (int c = 0; c < 25; ++c) {
                vals[c] = 32768.0f * xf[j + 16 * c];
                s += vals[c];
            }
        }
        // reduce over the 16-lane group (xor masks 1,2,4,8 stay within 16 lanes)
        #pragma unroll
        for (int off = 1; off < 16; off <<= 1) s += __shfl_xor(s, off, 32);
        const float mean = s * (1.0f / (float)FRAME_LEN);

        if (valid) {
            #pragma unroll
            for (int c = 0; c < 25; ++c) {
                const int n = j + 16 * c;
                const float v  = vals[c] - mean;
                const float pv = (n == 0) ? v : (32768.0f * xf[n - 1] - mean);
                frames[frl * ROWPAD + n] = (v - 0.97f * pv) * wnd[n];
            }
        } else {
            #pragma unroll
            for (int c = 0; c < 25; ++c) frames[frl * ROWPAD + j + 16 * c] = 0.0f;
        }
    }
    __syncthreads();

    // ---- phase 2: real DFT power spectrum via fp32 WMMA GEMM ----------------
    // D(16 frames x 16 bins) += A(16x4 samples) * B(4x16 twiddles), K = 400
    // (window length; zero-padded tail contributes nothing).
    // A layout (ISA 7.12.2, 32-bit A 16x4): lanes 0-15 hold M=lane, K=0/1 in
    // VGPR0/1; lanes 16-31 hold K=2/3.  B rows K=0..3 follow the same half-wave
    // split.  Each wave computes cos+sin accumulators for two 16-bin tiles.
    {
        const int col  = lane & 15;                // N within tile / A frame row
        const int half = lane >> 4;                // 0: K=0,1   1: K=2,3
        const int bin0 = (2 * wave) * 16 + col;    // tile 2w
        const int bin1 = bin0 + 16;                // tile 2w+1
        const float* arow = &frames[col * ROWPAD + 2 * half];

        unsigned i0 = ((unsigned)(bin0 * 2 * half)) & 511u;
        unsigned i1 = ((unsigned)(bin1 * 2 * half)) & 511u;

        v8f accC0 = {}, accS0 = {}, accC1 = {}, accS1 = {};

        #pragma unroll 4
        for (int k = 0; k < KMAX; k += 4) {
            const v2f a = *(const v2f*)(arow + k);          // ds_load_b64

            const float2 tA0 = trig[i0];
            const float2 tB0 = trig[(i0 + (unsigned)bin0) & 511u];
            const float2 tA1 = trig[i1];
            const float2 tB1 = trig[(i1 + (unsigned)bin1) & 511u];

            v2f bc0; bc0.x = tA0.x; bc0.y = tB0.x;
            v2f bs0; bs0.x = tA0.y; bs0.y = tB0.y;
            v2f bc1; bc1.x = tA1.x; bc1.y = tB1.x;
            v2f bs1; bs1.x = tA1.y; bs1.y = tB1.y;

            accC0 = __builtin_amdgcn_wmma_f32_16x16x4_f32(false, a, false, bc0,
                                                          (short)0, accC0, false, false);
            accS0 = __builtin_amdgcn_wmma_f32_16x16x4_f32(false, a, false, bs0,
                                                          (short)0, accS0, false, false);
            accC1 = __builtin_amdgcn_wmma_f32_16x16x4_f32(false, a, false, bc1,
                                                          (short)0, accC1, false, false);
            accS1 = __builtin_amdgcn_wmma_f32_16x16x4_f32(false, a, false, bs1,
                                                          (short)0, accS1, false, false);

            i0 = (i0 + (unsigned)(4 * bin0)) & 511u;
            i1 = (i1 + (unsigned)(4 * bin1)) & 511u;
        }

        // power = Re^2 + Im^2 ; C/D layout: VGPR r -> M = r + 8*half, N = col
        #pragma unroll
        for (int r = 0; r < 8; ++r) {
            const int M = r + 8 * half;
            const float c0 = accC0[r], s0 = accS0[r];
            const float c1 = accC1[r], s1 = accS1[r];
            spec[M * SPECPAD + bin0] = c0 * c0 + s0 * s0;
            spec[M * SPECPAD + bin1] = c1 * c1 + s1 * s1;
        }
    }
    __syncthreads();

    // ---- phase 3: mel projection + log, band-limited per filter ------------
    // Each mel triangle spans only ~6-8 FFT bins; invert the mel map to get
    // per-filter bin bounds (+-1 margin; the max(0,.) clamp keeps exactness).
    {
        const float mlow   = 1127.0f * log1pf(20.0f / 700.0f);
        const float mhigh  = 1127.0f * log1pf(8000.0f / 700.0f);
        const float delta  = (mhigh - mlow) * (1.0f / (float)(NMEL + 1));
        const float invd   = 1.0f / delta;

        for (int o = tid; o < FR_TILE * NMEL; o += 256) {
            const int frl = o / NMEL;
            const int m   = o - frl * NMEL;
            const int frg = fr0 + frl;
            if (frg >= NFRAMES) continue;

            const float left  = mlow + (float)m * delta;
            const float right = left + 2.0f * delta;

            // inverse mel -> Hz -> bin index bounds
            const float fl = 700.0f * (expf(left  * (1.0f / 1127.0f)) - 1.0f);
            const float fh = 700.0f * (expf(right * (1.0f / 1127.0f)) - 1.0f);
            int klo = (int)floorf(fl * (1.0f / BINW));          // floor == ceil-1 margin
            int khi = (int)ceilf (fh * (1.0f / BINW)) + 1;
            klo = klo < 0 ? 0 : klo;
            khi = khi > NBINS - 1 ? NBINS - 1 : khi;

            const float* srow = &spec[frl * SPECPAD];
            float acc = 0.0f;
            for (int k = klo; k <= khi; ++k) {
                const float mk = melbin[k];
                float w = fminf(mk - left, right - mk) * invd;   // tri filter
                w = fmaxf(w, 0.0f);
                acc = fmaf(srow[k], w, acc);
            }
            out[((size_t)batch * NFRAMES + frg) * NMEL + m] =
                logf(fmaxf(acc, 1.1920929e-07f));
        }
    }
}

extern "C" void kernel_launch(void* const* d_in, const int* in_sizes, int n_in,
                              void* d_out, int out_size, void* d_ws, size_t ws_size,
                              hipStream_t stream)
{
    (void)in_sizes; (void)n_in; (void)d_ws; (void)ws_size; (void)out_size;
    const float* x = (const float*)d_in[0];
    float* out = (float*)d_out;
    fbank_kernel<<<dim3(BATCH * NTILES), dim3(256), 0, stream>>>(x, out);
}